// RGCN_6468220747932
// MI455X (gfx1250) — compile-verified
//
#include <hip/hip_runtime.h>

// ---------------- problem constants (match reference) ----------------
namespace {
constexpr int NPn = 200000;   // paper nodes
constexpr int NAn = 200000;   // author nodes
constexpr int En  = 1000000;  // edges per relation
constexpr int D   = 128;      // DIN == DH (feature width of every GEMM A block)
constexpr int DO  = 64;       // DOUT
}

typedef __attribute__((ext_vector_type(2))) float v2f;
typedef __attribute__((ext_vector_type(8))) float v8f;

// ---------------- degree / inverse-degree ----------------
__global__ void degree_kernel(const int* __restrict__ dst, float* __restrict__ deg, int E) {
  int e = blockIdx.x * blockDim.x + threadIdx.x;
  if (e < E) atomicAdd(&deg[dst[e]], 1.0f);
}

__global__ void invdeg_kernel(float* __restrict__ deg, int n) {
  int i = blockIdx.x * blockDim.x + threadIdx.x;
  if (i < n) deg[i] = 1.0f / fmaxf(deg[i], 1.0f);
}

// ---------------- edge scatter-add: acc[dst] += X[src]  (one wave per edge) ----------------
__global__ void edge_scatter_kernel(const float* __restrict__ X,
                                    const int* __restrict__ src,
                                    const int* __restrict__ dst,
                                    float* __restrict__ acc, int E) {
  int e = blockIdx.x * (blockDim.x >> 5) + (threadIdx.x >> 5);
  if (e >= E) return;
  int lane = threadIdx.x & 31;
  int s = __builtin_amdgcn_readfirstlane(src[e]);   // wave-uniform -> scalar addr math
  int d = __builtin_amdgcn_readfirstlane(dst[e]);
  const float4* xs = (const float4*)(X + (size_t)s * D);
  float4 v = xs[lane];                               // 32 lanes x float4 = 128 feats
  float* a = acc + (size_t)d * D + lane * 4;
  atomicAdd(a + 0, v.x);
  atomicAdd(a + 1, v.y);
  atomicAdd(a + 2, v.z);
  atomicAdd(a + 3, v.w);
}

// ---------------- one K=128 block of the WMMA f32 16x16x4 GEMM, 16x32 output tile ----------
// Compile-time N => all W offsets are immediates (no s_mul_u64 address chains);
// one A fragment (b64 load) feeds two WMMAs (two 16-wide N-subtiles).
template <int N>
__device__ __forceinline__ void gemm_block2(v8f& c0, v8f& c1,
                                            const float* __restrict__ A,
                                            const float* __restrict__ W,
                                            int arow, int col0, int lane) {
  const int kh = lane >> 4;      // 0 for lanes 0-15, 1 for lanes 16-31
  const int nl = lane & 15;
  const float* Ap = A + (size_t)arow * D + 2 * kh;           // A[row][k + 2h + j]
  const float* Wp = W + (size_t)(2 * kh) * N + col0 + nl;    // W[k + 2h + j][col]
#pragma unroll 8
  for (int k = 0; k < D; k += 4) {
    v2f a, b0, b1;
    a.x  = Ap[k];
    a.y  = Ap[k + 1];
    b0.x = Wp[k * N];
    b0.y = Wp[(k + 1) * N];
    b1.x = Wp[k * N + 16];
    b1.y = Wp[(k + 1) * N + 16];
    c0 = __builtin_amdgcn_wmma_f32_16x16x4_f32(false, a, false, b0,
                                               (short)0, c0, false, false);
    c1 = __builtin_amdgcn_wmma_f32_16x16x4_f32(false, a, false, b1,
                                               (short)0, c1, false, false);
  }
}

// ---------------- fused multi-relation GEMM:
// Out = A0@W0 + diag(S1)*(A1@W1) + diag(S2)*(A2@W2) + bias, optional ReLU ------------
template <int N, bool RELU>
__global__ void rgcn_gemm_kernel(const float* __restrict__ A0,
                                 const float* __restrict__ A1, const float* __restrict__ S1,
                                 const float* __restrict__ A2, const float* __restrict__ S2,
                                 const float* __restrict__ W0, const float* __restrict__ W1,
                                 const float* __restrict__ W2,
                                 const float* __restrict__ bias,
                                 float* __restrict__ Out, int M) {
  constexpr int NT = N / 32;                                   // 16x32 tiles per row-block
  const int lane = threadIdx.x & 31;
  const int wave = threadIdx.x >> 5;
  const int tile = blockIdx.x * (blockDim.x >> 5) + wave;      // wave-uniform
  const int ntiles = (M >> 4) * NT;
  if (tile >= ntiles) return;                                  // whole-wave exit: EXEC stays full
  const int tm = tile / NT;
  const int tn = tile - tm * NT;
  const int row0 = tm << 4, col0 = tn << 5;
  const int nl = lane & 15, kh = lane >> 4;
  const int arow = row0 + nl;

  v8f acc0 = {}, acc1 = {};
  gemm_block2<N>(acc0, acc1, A0, W0, arow, col0, lane);        // root term (no scale)

  if (A1) {
    v8f t0 = {}, t1 = {};
    gemm_block2<N>(t0, t1, A1, W1, arow, col0, lane);
#pragma unroll
    for (int r = 0; r < 8; ++r) {
      const float s = S1[row0 + r + 8 * kh];                   // fold 1/deg into epilogue
      acc0[r] += t0[r] * s;
      acc1[r] += t1[r] * s;
    }
  }
  if (A2) {
    v8f t0 = {}, t1 = {};
    gemm_block2<N>(t0, t1, A2, W2, arow, col0, lane);
#pragma unroll
    for (int r = 0; r < 8; ++r) {
      const float s = S2[row0 + r + 8 * kh];
      acc0[r] += t0[r] * s;
      acc1[r] += t1[r] * s;
    }
  }

  const float bv0 = bias[col0 + nl];
  const float bv1 = bias[col0 + 16 + nl];
#pragma unroll
  for (int r = 0; r < 8; ++r) {
    float u = acc0[r] + bv0;
    float v = acc1[r] + bv1;
    if (RELU) { u = fmaxf(u, 0.0f); v = fmaxf(v, 0.0f); }
    const size_t rb = (size_t)(row0 + r + 8 * kh) * N;         // C/D layout: row r+8h, col nl
    Out[rb + col0 + nl]      = u;
    Out[rb + col0 + 16 + nl] = v;
  }
}

// ---------------- host orchestration ----------------
extern "C" void kernel_launch(void* const* d_in, const int* in_sizes, int n_in,
                              void* d_out, int out_size, void* d_ws, size_t ws_size,
                              hipStream_t stream) {
  (void)in_sizes; (void)n_in; (void)out_size;

  const float* x_paper    = (const float*)d_in[0];
  const float* emb_author = (const float*)d_in[1];
  const float* w_root_p1  = (const float*)d_in[2];
  const float* b_root_p1  = (const float*)d_in[3];
  const float* w_root_a1  = (const float*)d_in[4];
  const float* b_root_a1  = (const float*)d_in[5];
  const float* w_r1_1     = (const float*)d_in[6];
  const float* w_r2_1     = (const float*)d_in[7];
  const float* w_r3_1     = (const float*)d_in[8];
  const float* w_root_p2  = (const float*)d_in[9];
  const float* b_root_p2  = (const float*)d_in[10];
  const float* w_root_a2  = (const float*)d_in[11];
  const float* b_root_a2  = (const float*)d_in[12];
  const float* w_r1_2     = (const float*)d_in[13];
  const float* w_r2_2     = (const float*)d_in[14];
  const float* w_r3_2     = (const float*)d_in[15];
  const int* e1_src = (const int*)d_in[16];
  const int* e1_dst = (const int*)d_in[17];
  const int* e2_src = (const int*)d_in[18];
  const int* e2_dst = (const int*)d_in[19];
  const int* e3_src = (const int*)d_in[20];
  const int* e3_dst = (const int*)d_in[21];

  // workspace carve-out (256B aligned; sizes chosen so buffers are contiguous)
  size_t off = 0;
  auto take = [&](size_t nfloat) {
    size_t p = off;
    off += ((nfloat * sizeof(float)) + 255) & ~(size_t)255;
    return (float*)((char*)d_ws + p);
  };
  float* aggA = take((size_t)NPn * D);   // relation 1 sums (author -> paper)
  float* aggB = take((size_t)NPn * D);   // relation 2 sums (paper  -> paper)
  float* aggC = take((size_t)NAn * D);   // relation 3 sums (paper  -> author)
  float* hp   = take((size_t)NPn * D);   // hidden paper
  float* ha   = take((size_t)NAn * D);   // hidden author
  float* inv1 = take(NPn);               // 1/max(deg,1) per relation
  float* inv2 = take(NPn);
  float* inv3 = take(NAn);
  if (off > ws_size) return;             // workspace too small: bail safely

  const size_t aggBytes = (size_t)(2 * NPn + NAn) * D * sizeof(float);
  const size_t degBytes = (size_t)(2 * NPn + NAn) * sizeof(float);
  hipMemsetAsync(aggA, 0, aggBytes, stream);   // aggA..aggC contiguous
  hipMemsetAsync(inv1, 0, degBytes, stream);   // inv1..inv3 contiguous

  const int tb = 256;
  // degrees (edge structure is layer-invariant -> compute once)
  degree_kernel<<<(En + tb - 1) / tb, tb, 0, stream>>>(e1_dst, inv1, En);
  degree_kernel<<<(En + tb - 1) / tb, tb, 0, stream>>>(e2_dst, inv2, En);
  degree_kernel<<<(En + tb - 1) / tb, tb, 0, stream>>>(e3_dst, inv3, En);
  invdeg_kernel<<<(NPn + tb - 1) / tb, tb, 0, stream>>>(inv1, NPn);
  invdeg_kernel<<<(NPn + tb - 1) / tb, tb, 0, stream>>>(inv2, NPn);
  invdeg_kernel<<<(NAn + tb - 1) / tb, tb, 0, stream>>>(inv3, NAn);

  // ---- layer 1 aggregation (raw inputs) ----
  const int sblocks = En / 8;            // 8 waves per 256-thread block, 1 wave per edge
  edge_scatter_kernel<<<sblocks, 256, 0, stream>>>(emb_author, e1_src, e1_dst, aggA, En);
  edge_scatter_kernel<<<sblocks, 256, 0, stream>>>(x_paper,    e2_src, e2_dst, aggB, En);
  edge_scatter_kernel<<<sblocks, 256, 0, stream>>>(x_paper,    e3_src, e3_dst, aggC, En);

  // ---- layer 1 fused GEMMs (+bias, ReLU) ----
  {
    const int tilesP = (NPn / 16) * (D / 32);
    rgcn_gemm_kernel<D, true><<<(tilesP + 7) / 8, 256, 0, stream>>>(
        x_paper, aggA, inv1, aggB, inv2,
        w_root_p1, w_r1_1, w_r2_1, b_root_p1, hp, NPn);
    const int tilesA = (NAn / 16) * (D / 32);
    rgcn_gemm_kernel<D, true><<<(tilesA + 7) / 8, 256, 0, stream>>>(
        emb_author, aggC, inv3, nullptr, nullptr,
        w_root_a1, w_r3_1, nullptr, b_root_a1, ha, NAn);
  }

  // ---- layer 2 aggregation (hidden states) ----
  hipMemsetAsync(aggA, 0, aggBytes, stream);
  edge_scatter_kernel<<<sblocks, 256, 0, stream>>>(ha, e1_src, e1_dst, aggA, En);
  edge_scatter_kernel<<<sblocks, 256, 0, stream>>>(hp, e2_src, e2_dst, aggB, En);
  edge_scatter_kernel<<<sblocks, 256, 0, stream>>>(hp, e3_src, e3_dst, aggC, En);

  // ---- layer 2 fused GEMMs straight into d_out ----
  float* out_p = (float*)d_out;
  float* out_a = out_p + (size_t)NPn * DO;
  {
    const int tilesP = (NPn / 16) * (DO / 32);
    rgcn_gemm_kernel<DO, false><<<(tilesP + 7) / 8, 256, 0, stream>>>(
        hp, aggA, inv1, aggB, inv2,
        w_root_p2, w_r1_2, w_r2_2, b_root_p2, out_p, NPn);
    const int tilesA = (NAn / 16) * (DO / 32);
    rgcn_gemm_kernel<DO, false><<<(tilesA + 7) / 8, 256, 0, stream>>>(
        ha, aggC, inv3, nullptr, nullptr,
        w_root_a2, w_r3_2, nullptr, b_root_a2, out_a, NAn);
  }
}